// Attention_talking_head_2010044694917
// MI455X (gfx1250) — compile-verified
//
#include <hip/hip_runtime.h>
#include <hip/hip_bf16.h>

// MI455X / gfx1250, wave32. Talking-heads attention, all-fp32 WMMA path,
// with CDNA5 async-to-LDS staging and TDM tensor_load_to_lds.

#define N_TOK 245
#define NBKT  1698
#define SCALE 0.125f   // 64^-0.5

typedef __attribute__((ext_vector_type(2))) float    v2f;
typedef __attribute__((ext_vector_type(4))) float    v4f;
typedef __attribute__((ext_vector_type(8))) float    v8f;
typedef __attribute__((ext_vector_type(4))) unsigned v4u;
typedef __attribute__((ext_vector_type(4))) int      v4i;
typedef __attribute__((ext_vector_type(8))) int      v8i;

static __device__ __forceinline__ v8f wmma_f32(v2f a, v2f b, v8f c) {
  // (neg_a, A, neg_b, B, c_mod, C, reuse_a, reuse_b)
  return __builtin_amdgcn_wmma_f32_16x16x4_f32(false, a, false, b, (short)0, c,
                                               false, false);
}

// Async copy 16B global -> LDS (per lane), tracked with ASYNCcnt.
static __device__ __forceinline__ void async_g2l_b128(unsigned lds_byte,
                                                      const void* gaddr) {
  asm volatile("global_load_async_to_lds_b128 %0, %1, off"
               :: "v"(lds_byte), "v"(gaddr) : "memory");
}
static __device__ __forceinline__ void wait_async0() {
  asm volatile("s_wait_asynccnt 0x0" ::: "memory");
}

// ---------------------------------------------------------------------------
// Kernel 0: expand RPE table -> biasT[H][245][245]
// ---------------------------------------------------------------------------
__global__ void bias_expand(const float* __restrict__ rpe,
                            const int* __restrict__ rel,
                            float* __restrict__ biasT) {
  int idx = blockIdx.x * 256 + threadIdx.x;
  if (idx >= 12 * N_TOK * N_TOK) return;
  int h = idx / (N_TOK * N_TOK);
  int r = idx - h * (N_TOK * N_TOK);
  biasT[idx] = rpe[h * NBKT + rel[r]];
}

// ---------------------------------------------------------------------------
// Kernel 1: QKV projection. X (15680x768) @ Wqkv^T (768x2304).
//   Q   : (B,H,256,64)          scaled by SCALE
//   Kt2 : (B,H) x [(d>>1)*256 + m]*2 + (d&1)   (d-paired transpose, 16384/head)
//   Vp  : (B,H) x [(m>>1)*64 + d]*2 + (m&1)    (m-paired,           16384/head)
// grid (18, 980), block 256 (8 waves; wave = one 16x16 output tile)
// ---------------------------------------------------------------------------
__launch_bounds__(256, 2)
__global__ void qkv_gemm(const float* __restrict__ x, const float* __restrict__ w,
                         float* __restrict__ Q, float* __restrict__ Kt2,
                         float* __restrict__ Vp) {
  __shared__ float shx[16 * 772];  // stride 772 -> bank = 4*row + k (no conflicts)
  const int tid = threadIdx.x;
  const int rt  = blockIdx.y;
  {
    unsigned lbase = (unsigned)(uintptr_t)shx;
    for (int idx = tid * 4; idx < 16 * 768; idx += 1024) {
      int i = idx / 768, col = idx - i * 768;
      async_g2l_b128(lbase + (unsigned)(i * 772 + col) * 4u,
                     x + (size_t)(rt * 16 + i) * 768 + col);
    }
    wait_async0();
  }
  __syncthreads();

  const int wid = tid >> 5, lane = tid & 31, lo = lane & 15, hi = lane >> 4;
  const int ct = blockIdx.x * 8 + wid;   // 0..143
  const int ebase = ct * 16;
  const float* wrow = w + (size_t)(ebase + lo) * 768 + 2 * hi;
  const float* arow = shx + lo * 772 + 2 * hi;
  v8f acc; for (int i = 0; i < 8; ++i) acc[i] = 0.f;
  for (int k = 0; k < 768; k += 4) {
    v2f a  = *(const v2f*)(arow + k);
    v2f bm = *(const v2f*)(wrow + k);
    acc = wmma_f32(a, bm, acc);
  }
  const int e = ebase + lo;
  const int s = e / 768;
  const int rem = e - s * 768;
  const int h = rem >> 6, d = rem & 63;
  for (int j = 0; j < 8; ++j) {
    int r = rt * 16 + j + 8 * hi;        // flat row < 15680 always
    int b = r / 245, n = r - b * 245;
    float val = acc[j];
    size_t bh = (size_t)(b * 12 + h);
    if (s == 0)
      Q[(bh * 256 + n) * 64 + d] = val * SCALE;
    else if (s == 1)
      Kt2[bh * 16384 + ((size_t)(d >> 1) * 256 + n) * 2 + (d & 1)] = val;
    else
      Vp[bh * 16384 + ((size_t)(n >> 1) * 64 + d) * 2 + (n & 1)] = val;
  }
}

// ---------------------------------------------------------------------------
// Kernel 2: fused attention. One block per (b, 16-row n-tile).
// LDS: sQ[12][16][68] (filled by TDM with hardware padding), sS[12][16][260].
// 512 threads (16 waves).
// ---------------------------------------------------------------------------
__launch_bounds__(512, 1)
__global__ void attn_fused(const float* __restrict__ Q, const float* __restrict__ Kt2,
                           const float* __restrict__ Vp, const float* __restrict__ biasT,
                           const float* __restrict__ w_l, const float* __restrict__ b_l,
                           const float* __restrict__ w_w, const float* __restrict__ b_w,
                           float* __restrict__ O) {
  extern __shared__ float smem[];
  float* sQ  = smem;            // 12*16*68  = 13056
  float* sS  = smem + 13056;    // 12*16*260 = 49920
  float* sWl = sS + 49920;      // 144
  float* sWw = sWl + 144;       // 144
  float* sBl = sWw + 144;       // 12
  float* sBw = sBl + 12;        // 12

  const int tid = threadIdx.x;
  const int b   = blockIdx.x >> 4;
  const int n0  = (blockIdx.x & 15) * 16;
  const size_t bh0 = (size_t)b * 12;
  const int wid = tid >> 5, lane = tid & 31, lo = lane & 15, hi = lane >> 4;

  // Phase 1: TDM-stage Q tile (one 16x64 fp32 tile per head, waves 0..11).
  // D# pad feature inserts 4 DWORDs after every 64 DWORDs -> LDS stride 68.
  if (wid < 12) {
    int h = __builtin_amdgcn_readfirstlane(wid);
    unsigned lds_addr = (unsigned)(uintptr_t)sQ + (unsigned)h * 4352u; // bytes
    unsigned long long ga =
        (unsigned long long)(uintptr_t)(Q + ((bh0 + h) * 256 + n0) * 64);
    v4u g0; v8i g1; v4i g2; v4i g3; v8i g4;
    g0[0] = 1u;                                    // count=1 (valid descriptor)
    g0[1] = lds_addr;                              // lds_addr (bytes)
    g0[2] = (unsigned)(ga & 0xffffffffu);          // global_addr[31:0]
    g0[3] = (unsigned)((ga >> 32) & 0x01ffffffu)   // global_addr[56:32]
          | 0x80000000u;                           // type = 2 ("image")
    g1[0] = (int)((2u << 16)                       // data_size = 4 bytes
                | (1u << 20)                       // pad_enable
                | (5u << 22)                       // pad_interval: 64 DWORDs
                | (3u << 25));                     // pad_amount:   4 DWORDs
    g1[1] = (int)(64u << 16);                      // tensor_dim0 = 64
    g1[2] = (int)(16u << 16);                      // tensor_dim1 = 16
    g1[3] = (int)(64u << 16);                      // tile_dim0   = 64
    g1[4] = 16;                                    // tile_dim1   = 16
    g1[5] = 64;                                    // tensor_dim0_stride = 64
    g1[6] = 0; g1[7] = 0;
    g2[0] = 0; g2[1] = 0; g2[2] = 0; g2[3] = 0;
    g3[0] = 0; g3[1] = 0; g3[2] = 0; g3[3] = 0;
    g4[0] = 0; g4[1] = 0; g4[2] = 0; g4[3] = 0;
    g4[4] = 0; g4[5] = 0; g4[6] = 0; g4[7] = 0;
    __builtin_amdgcn_tensor_load_to_lds(g0, g1, g2, g3, g4, 0);
    __builtin_amdgcn_s_wait_tensorcnt(0);
  }
  if (tid < 144) { sWl[tid] = w_l[tid]; sWw[tid] = w_w[tid]; }
  if (tid < 12)  { sBl[tid] = b_l[tid]; sBw[tid] = b_w[tid]; }
  __syncthreads();

  // Phase 2: S[h] = Q[h] @ K[h]^T   (12 heads x 16 m-tiles, K=64)
  for (int t = wid; t < 192; t += 16) {
    int h = t >> 4, mt = t & 15;
    const float* kt = Kt2 + (bh0 + h) * 16384 + (size_t)(mt * 16 + lo) * 2;
    const float* aq = sQ + h * 1088 + lo * 68 + 2 * hi;
    v8f acc; for (int i = 0; i < 8; ++i) acc[i] = 0.f;
    for (int k = 0; k < 64; k += 4) {
      v2f a  = *(const v2f*)(aq + k);
      v2f bm = *(const v2f*)(kt + ((k >> 1) + hi) * 512);
      acc = wmma_f32(a, bm, acc);
    }
    float* srow = sS + h * 4160 + mt * 16 + lo;
    for (int j = 0; j < 8; ++j) srow[(j + 8 * hi) * 260] = acc[j];
  }
  __syncthreads();

  // Phase 3: + RPE bias, pre-softmax talking heads (w_l), mask pad columns
  for (int p = tid; p < 4096; p += 512) {
    int n = p >> 8, m = p & 255;
    int nn = min(n0 + n, 244);
    float vh[12];
    if (m < N_TOK) {
      const float* bp = biasT + nn * 245 + m;
      for (int h = 0; h < 12; ++h)
        vh[h] = sS[h * 4160 + n * 260 + m] + bp[h * 60025];
    } else {
      for (int h = 0; h < 12; ++h) vh[h] = 0.f;
    }
    for (int g = 0; g < 12; ++g) {
      float o;
      if (m < N_TOK) {
        o = sBl[g];
        for (int h = 0; h < 12; ++h) o = fmaf(vh[h], sWl[g * 12 + h], o);
      } else {
        o = -1e30f;
      }
      sS[g * 4160 + n * 260 + m] = o;
    }
  }
  __syncthreads();

  // Phase 4: softmax over m (one wave32 per (g,n) row of 256)
  for (int row = wid; row < 192; row += 16) {
    float* base = sS + (row >> 4) * 4160 + (row & 15) * 260;
    float r[8], mx = -3.4e38f;
    for (int i = 0; i < 8; ++i) { r[i] = base[lane + 32 * i]; mx = fmaxf(mx, r[i]); }
    for (int off = 16; off > 0; off >>= 1) mx = fmaxf(mx, __shfl_xor(mx, off, 32));
    float sum = 0.f;
    for (int i = 0; i < 8; ++i) { r[i] = __expf(r[i] - mx); sum += r[i]; }
    for (int off = 16; off > 0; off >>= 1) sum += __shfl_xor(sum, off, 32);
    float inv = 1.0f / sum;
    for (int i = 0; i < 8; ++i) base[lane + 32 * i] = r[i] * inv;
  }
  __syncthreads();

  // Phase 5: post-softmax talking heads (w_w); pad columns -> exactly 0
  for (int p = tid; p < 4096; p += 512) {
    int n = p >> 8, m = p & 255;
    float vh[12];
    for (int h = 0; h < 12; ++h) vh[h] = sS[h * 4160 + n * 260 + m];
    for (int g = 0; g < 12; ++g) {
      float o = 0.f;
      if (m < N_TOK) {
        o = sBw[g];
        for (int h = 0; h < 12; ++h) o = fmaf(vh[h], sWw[g * 12 + h], o);
      }
      sS[g * 4160 + n * 260 + m] = o;
    }
  }
  __syncthreads();

  // Phase 6: O[g] = attn2[g] @ V[g]   (12 heads x 4 d-tiles, K=256)
  for (int t = wid; t < 48; t += 16) {
    int h = t >> 2, dt = t & 3;
    const float* vb = Vp + (bh0 + h) * 16384 + (size_t)(dt * 16 + lo) * 2;
    const float* as = sS + h * 4160 + lo * 260 + 2 * hi;
    v8f acc; for (int i = 0; i < 8; ++i) acc[i] = 0.f;
    for (int k = 0; k < 256; k += 4) {
      v2f a  = *(const v2f*)(as + k);
      v2f bm = *(const v2f*)(vb + ((k >> 1) + hi) * 128);
      acc = wmma_f32(a, bm, acc);
    }
    for (int j = 0; j < 8; ++j) {
      int ng = n0 + j + 8 * hi;
      if (ng < N_TOK)
        O[(size_t)(b * 245 + ng) * 768 + h * 64 + dt * 16 + lo] = acc[j];
    }
  }
}

// ---------------------------------------------------------------------------
// Kernel 3: output projection. O (15680x768) @ Wproj^T (768x768) + b
// grid (6, 980), block 256
// ---------------------------------------------------------------------------
__launch_bounds__(256, 2)
__global__ void proj_gemm(const float* __restrict__ Oin, const float* __restrict__ w,
                          const float* __restrict__ bias, float* __restrict__ out) {
  __shared__ float sho[16 * 772];
  const int tid = threadIdx.x;
  const int rt  = blockIdx.y;
  {
    unsigned lbase = (unsigned)(uintptr_t)sho;
    for (int idx = tid * 4; idx < 16 * 768; idx += 1024) {
      int i = idx / 768, col = idx - i * 768;
      async_g2l_b128(lbase + (unsigned)(i * 772 + col) * 4u,
                     Oin + (size_t)(rt * 16 + i) * 768 + col);
    }
    wait_async0();
  }
  __syncthreads();

  const int wid = tid >> 5, lane = tid & 31, lo = lane & 15, hi = lane >> 4;
  const int obase = (blockIdx.x * 8 + wid) * 16;
  const float* wrow = w + (size_t)(obase + lo) * 768 + 2 * hi;
  const float* arow = sho + lo * 772 + 2 * hi;
  v8f acc; for (int i = 0; i < 8; ++i) acc[i] = 0.f;
  for (int k = 0; k < 768; k += 4) {
    v2f a  = *(const v2f*)(arow + k);
    v2f bm = *(const v2f*)(wrow + k);
    acc = wmma_f32(a, bm, acc);
  }
  float bv = bias[obase + lo];
  for (int j = 0; j < 8; ++j) {
    int r = rt * 16 + j + 8 * hi;
    out[(size_t)r * 768 + obase + lo] = acc[j] + bv;
  }
}

// ---------------------------------------------------------------------------
extern "C" void kernel_launch(void* const* d_in, const int* in_sizes, int n_in,
                              void* d_out, int out_size, void* d_ws, size_t ws_size,
                              hipStream_t stream) {
  (void)in_sizes; (void)n_in; (void)out_size; (void)ws_size;
  const float* x      = (const float*)d_in[0];
  const float* w_qkv  = (const float*)d_in[1];
  const float* w_proj = (const float*)d_in[2];
  const float* b_proj = (const float*)d_in[3];
  const float* w_l    = (const float*)d_in[4];
  const float* b_l    = (const float*)d_in[5];
  const float* w_w    = (const float*)d_in[6];
  const float* b_w    = (const float*)d_in[7];
  const float* rpe    = (const float*)d_in[8];
  const int*   rel    = (const int*)d_in[9];
  float* out = (float*)d_out;
  float* ws  = (float*)d_ws;

  float* Q     = ws;                    // (B,H,256,64)         12,582,912 f
  float* Kt2   = Q + 12582912;          // d-paired K transpose 12,582,912 f
  float* Vp    = Kt2 + 12582912;        // m-paired V           12,582,912 f
  float* biasT = Vp + 12582912;         // (H,245,245)             720,300 f
  float* O     = biasT + 720300;        // (B*245, 768)         12,042,240 f

  bias_expand<<<(12 * N_TOK * N_TOK + 255) / 256, 256, 0, stream>>>(rpe, rel, biasT);
  qkv_gemm<<<dim3(18, 980), 256, 0, stream>>>(x, w_qkv, Q, Kt2, Vp);
  size_t smem = (size_t)(13056 + 49920 + 144 + 144 + 12 + 12) * sizeof(float); // 253,152 B
  attn_fused<<<1024, 512, smem, stream>>>(Q, Kt2, Vp, biasT, w_l, b_l, w_w, b_w, O);
  proj_gemm<<<dim3(6, 980), 256, 0, stream>>>(O, w_proj, b_proj, out);
}